// WeightedLossWithPrior_9723805958292
// MI455X (gfx1250) — compile-verified
//
#include <hip/hip_runtime.h>

// CDNA5 / gfx1250, wave32.
// Masked BCE-with-logits mean:  sum(bce*m)/max(sum(m),1)
// Memory-bound: 192 MB @ 23.3 TB/s ~= 8.2 us floor. One streaming pass with
// b128 loads, per-lane fp32 accumulation, wave reduction through the f32 WMMA
// pipe (A-side), deterministic two-pass block reduction (no float atomics).

typedef float v2f __attribute__((ext_vector_type(2)));
typedef float v8f __attribute__((ext_vector_type(8)));

#define BCE_GRID 2048
#define BCE_BLK  256
#define WAVES_PER_BLK (BCE_BLK / 32)

// Wave-wide (32-lane) fp32 sum using V_WMMA_F32_16X16X4_F32, values on the
// A side:  A(16x4) = {lane value, 0}, B(4x16) = all-ones
//   => D[m][n] = sum_k A[m][k] = rowsum[m]   (independent of n)
// Per lane, the 8 D VGPRs hold 8 distinct rows; lanes L and L^16 hold the
// complementary 8 rows (each D element lives in exactly one (VGPR,lane) slot).
// So  (sum of 8 D regs)  +  (same from the other half-wave)  ==  full wave
// sum, under ANY row<->VGPR layout permutation. Epilogue cost: 1 WMMA,
// 7 VALU adds, 1 cross-half shuffle (vs. a serialized 4-step LDS butterfly).
__device__ __forceinline__ float wave_sum32(float v) {
  v2f a; a.x = v;    a.y = 0.0f;   // this lane's two A slots
  v2f b; b.x = 1.0f; b.y = 1.0f;   // all-ones B (4x16)
  v8f c = {};
  v8f d = __builtin_amdgcn_wmma_f32_16x16x4_f32(
      /*neg_a=*/false, a, /*neg_b=*/false, b,
      /*c_mod=*/(short)0, c, /*reuse_a=*/false, /*reuse_b=*/false);
  float s = ((d[0] + d[1]) + (d[2] + d[3])) + ((d[4] + d[5]) + (d[6] + d[7]));
  s += __shfl_xor(s, 16, 32);      // add the other half-wave's 8 rows
  return s;                        // full 32-lane total, in every lane
}

// Numerically stable BCE-with-logits: max(x,0) - x*t + log1p(exp(-|x|))
__device__ __forceinline__ float bce_logits(float x, float t) {
  float e = __expf(-__builtin_fabsf(x));          // v_exp_f32 (TRANS)
  return __builtin_fmaf(-x, t, __builtin_fmaxf(x, 0.0f)) +
         __logf(1.0f + e);                        // v_log_f32 (TRANS)
}

__global__ __launch_bounds__(BCE_BLK)
void bce_partial_kernel(const float* __restrict__ logits,
                        const float* __restrict__ targets,
                        const int*   __restrict__ mask,
                        float*       __restrict__ partials,
                        int n) {
  const int tid    = blockIdx.x * BCE_BLK + threadIdx.x;
  const int stride = gridDim.x * BCE_BLK;
  const int n4     = n >> 2;

  const float4* __restrict__ x4 = (const float4*)logits;
  const float4* __restrict__ t4 = (const float4*)targets;
  const int4*   __restrict__ m4 = (const int4*)mask;

  float s_bce = 0.0f;
  float s_cnt = 0.0f;

  for (int i = tid; i < n4; i += stride) {
    float4 x = x4[i];              // global_load_b128, fully coalesced
    float4 t = t4[i];
    int4   m = m4[i];
    float m0 = (float)m.x, m1 = (float)m.y, m2 = (float)m.z, m3 = (float)m.w;
    s_bce += bce_logits(x.x, t.x) * m0;
    s_bce += bce_logits(x.y, t.y) * m1;
    s_bce += bce_logits(x.z, t.z) * m2;
    s_bce += bce_logits(x.w, t.w) * m3;
    s_cnt += m0 + m1 + m2 + m3;
  }

  // Scalar tail (n % 4 != 0). Dead for N = 16M, kept for generality;
  // control flow reconverges before the WMMA reduction.
  if (blockIdx.x == 0 && threadIdx.x == 0) {
    for (int i = (n4 << 2); i < n; ++i) {
      float mm = (float)mask[i];
      s_bce += bce_logits(logits[i], targets[i]) * mm;
      s_cnt += mm;
    }
  }

  // EXEC is all-ones here (uniform control flow) -> WMMA legal.
  float wb = wave_sum32(s_bce);
  float wc = wave_sum32(s_cnt);

  __shared__ float lb[WAVES_PER_BLK];
  __shared__ float lc[WAVES_PER_BLK];
  const int lane = threadIdx.x & 31;
  const int wave = threadIdx.x >> 5;
  if (lane == 0) { lb[wave] = wb; lc[wave] = wc; }
  __syncthreads();
  if (threadIdx.x == 0) {
    float tb = 0.0f, tc = 0.0f;
#pragma unroll
    for (int w = 0; w < WAVES_PER_BLK; ++w) { tb += lb[w]; tc += lc[w]; }
    partials[2 * blockIdx.x + 0] = tb;
    partials[2 * blockIdx.x + 1] = tc;
  }
}

__global__ __launch_bounds__(BCE_BLK)
void bce_final_kernel(const float* __restrict__ partials,
                      float*       __restrict__ out,
                      int nblocks) {
  float tb = 0.0f, tc = 0.0f;
  for (int i = threadIdx.x; i < nblocks; i += BCE_BLK) {
    tb += partials[2 * i + 0];
    tc += partials[2 * i + 1];
  }
  // Reconverged; fixed-order deterministic tree from here down.
  tb = wave_sum32(tb);
  tc = wave_sum32(tc);

  __shared__ float lb[WAVES_PER_BLK];
  __shared__ float lc[WAVES_PER_BLK];
  const int lane = threadIdx.x & 31;
  const int wave = threadIdx.x >> 5;
  if (lane == 0) { lb[wave] = tb; lc[wave] = tc; }
  __syncthreads();
  if (threadIdx.x == 0) {
    float sb = 0.0f, sc = 0.0f;
#pragma unroll
    for (int w = 0; w < WAVES_PER_BLK; ++w) { sb += lb[w]; sc += lc[w]; }
    out[0] = sb / __builtin_fmaxf(sc, 1.0f);
  }
}

extern "C" void kernel_launch(void* const* d_in, const int* in_sizes, int n_in,
                              void* d_out, int out_size, void* d_ws, size_t ws_size,
                              hipStream_t stream) {
  (void)n_in; (void)out_size;
  const float* logits  = (const float*)d_in[0];
  const float* targets = (const float*)d_in[1];
  const int*   mask    = (const int*)d_in[2];
  float* out      = (float*)d_out;
  float* partials = (float*)d_ws;   // needs grid * 2 floats (16 KB at 2048)
  const int n = in_sizes[0];

  int grid = BCE_GRID;
  size_t need = (size_t)grid * 2u * sizeof(float);
  if (ws_size < need) {             // defensive clamp; still deterministic
    grid = (int)(ws_size / (2u * sizeof(float)));
    if (grid < 1) grid = 1;
  }

  bce_partial_kernel<<<grid, BCE_BLK, 0, stream>>>(logits, targets, mask, partials, n);
  bce_final_kernel<<<1, BCE_BLK, 0, stream>>>(partials, out, grid);
}